// ModularMoERouter_20220706029770
// MI455X (gfx1250) — compile-verified
//
#include <hip/hip_runtime.h>
#include <hip/hip_bf16.h>
#include <math.h>

// MoE soft-router: T=8192 tokens, D=1024, E=16 experts, H=2048.
// Bandwidth-bound (~340MB compulsory traffic, ~0.94 GFLOP). fp32 throughout;
// V_WMMA_F32_16X16X4_F32 used for the two token-dim GEMMs.

typedef __attribute__((ext_vector_type(2))) float v2f;
typedef __attribute__((ext_vector_type(8))) float v8f;

#define T_TOK 8192
#define DIM   1024
#define NEXP  16
#define HDIM  2048

// ---------------- workspace layout (float offsets) ----------------
#define WS_LOGITS   0          // 8192*16
#define WS_COMBINE  131072     // 8192*16
#define WS_COLMAX   262144     // 16
#define WS_COLSUM   262160     // 16
#define WS_USAGE    262176     // 16
#define WS_ENT      262192     // 1 (padded to 16)
#define WS_SLOTS    262208     // 16*1024
#define WS_HPRE     278592     // 16*2048
#define WS_HACT     311360     // 16*2048
#define WS_SLOTOUT  344128     // 16*1024

// ---------------- init accumulators ----------------
__global__ void k_init(float* __restrict__ slots, float* __restrict__ usage,
                       float* __restrict__ ent) {
    int i = blockIdx.x * 256 + threadIdx.x;
    if (i < NEXP * DIM) slots[i] = 0.f;
    if (i < NEXP) usage[i] = 0.f;
    if (i == 0) *ent = 0.f;
}

// ---------------- logits = x @ router_w^T  (WMMA f32 16x16x4) ----------------
// one wave per 16-token tile; N = 16 experts; K = D in steps of 4.
__global__ void k_logits_wmma(const float* __restrict__ x, const float* __restrict__ rw,
                              float* __restrict__ logits) {
    const int lane = threadIdx.x & 31;
    const int wave = threadIdx.x >> 5;
    const int tile = blockIdx.x * 8 + wave;          // 512 tiles
    const int t0   = tile * 16;
    const int m    = lane & 15;                       // A row (token) == B col (expert)
    const int kk   = (lane >> 4) << 1;                // K sub-offset 0 or 2
    const float* xrow = x  + (size_t)(t0 + m) * DIM + kk;
    const float* brow = rw + (size_t)m * DIM + kk;    // router_w[expert=m][k]
    v8f c = {};
    for (int k = 0; k < DIM; k += 4) {
        v2f a, b;
        a[0] = xrow[k]; a[1] = xrow[k + 1];
        b[0] = brow[k]; b[1] = brow[k + 1];
        c = __builtin_amdgcn_wmma_f32_16x16x4_f32(false, a, false, b, (short)0, c,
                                                  false, false);
    }
    const int rbase = t0 + ((lane >> 4) << 3);
    #pragma unroll
    for (int i = 0; i < 8; ++i)
        logits[(size_t)(rbase + i) * NEXP + m] = c[i];   // TEMP == 1.0
}

// ---------------- per-expert (column) softmax stats over all tokens ----------------
__global__ void k_colstats(const float* __restrict__ logits, float* __restrict__ colmax,
                           float* __restrict__ colsum) {
    __shared__ float s[256];
    const int e = blockIdx.x, tid = threadIdx.x;
    float mx = -3.402823466e38f;
    for (int t = tid; t < T_TOK; t += 256) mx = fmaxf(mx, logits[t * NEXP + e]);
    s[tid] = mx; __syncthreads();
    for (int off = 128; off > 0; off >>= 1) {
        if (tid < off) s[tid] = fmaxf(s[tid], s[tid + off]);
        __syncthreads();
    }
    const float cm = s[0]; __syncthreads();
    float sm = 0.f;
    for (int t = tid; t < T_TOK; t += 256) sm += expf(logits[t * NEXP + e] - cm);
    s[tid] = sm; __syncthreads();
    for (int off = 128; off > 0; off >>= 1) {
        if (tid < off) s[tid] += s[tid + off];
        __syncthreads();
    }
    if (tid == 0) { colmax[e] = cm; colsum[e] = s[0]; }
}

// ---------------- per-token (row) softmax -> combine, + usage/entropy ----------------
__global__ void k_combine(const float* __restrict__ logits, float* __restrict__ combine,
                          float* __restrict__ usage, float* __restrict__ ent_acc) {
    __shared__ float s_usage[NEXP];
    __shared__ float s_ent;
    const int tid = threadIdx.x;
    if (tid < NEXP) s_usage[tid] = 0.f;
    if (tid == 0) s_ent = 0.f;
    __syncthreads();
    const int t = blockIdx.x * 256 + tid;
    float v[NEXP];
    #pragma unroll
    for (int e = 0; e < NEXP; ++e) v[e] = logits[(size_t)t * NEXP + e];
    float mx = v[0];
    #pragma unroll
    for (int e = 1; e < NEXP; ++e) mx = fmaxf(mx, v[e]);
    float se = 0.f;
    #pragma unroll
    for (int e = 0; e < NEXP; ++e) { v[e] = expf(v[e] - mx); se += v[e]; }
    const float inv = 1.f / se;
    float ent = 0.f;
    #pragma unroll
    for (int e = 0; e < NEXP; ++e) {
        float cc = v[e] * inv;
        combine[(size_t)t * NEXP + e] = cc;
        ent -= cc * logf(cc + 1e-8f);
        atomicAdd(&s_usage[e], cc);
    }
    atomicAdd(&s_ent, ent);
    __syncthreads();
    if (tid < NEXP) atomicAdd(&usage[tid], s_usage[tid]);
    if (tid == 0) atomicAdd(ent_acc, s_ent);
}

// ---------------- slots[e,:] = sum_t dispatch[t,e] * x[t,:] ----------------
// grid (chunk=16, expert=16); 512-token chunk per block, full D via float4.
__global__ void k_slots(const float* __restrict__ x, const float* __restrict__ logits,
                        const float* __restrict__ colmax, const float* __restrict__ colsum,
                        float* __restrict__ slots) {
    __shared__ float s_disp[512];
    const int chunk = blockIdx.x, e = blockIdx.y, tid = threadIdx.x;
    const float cm = colmax[e];
    const float rcs = 1.f / colsum[e];
    for (int i = tid; i < 512; i += 256) {
        int t = chunk * 512 + i;
        s_disp[i] = expf(logits[(size_t)t * NEXP + e] - cm) * rcs;
    }
    __syncthreads();
    float4 acc = {0.f, 0.f, 0.f, 0.f};
    const float4* x4 = (const float4*)x;
    for (int i = 0; i < 512; ++i) {
        int t = chunk * 512 + i;
        float4 xv = x4[(size_t)t * (DIM / 4) + tid];
        float w = s_disp[i];
        acc.x += w * xv.x; acc.y += w * xv.y; acc.z += w * xv.z; acc.w += w * xv.w;
    }
    float* sp = slots + (size_t)e * DIM + tid * 4;
    atomicAdd(sp + 0, acc.x); atomicAdd(sp + 1, acc.y);
    atomicAdd(sp + 2, acc.z); atomicAdd(sp + 3, acc.w);
}

// ---------------- h_pre[e,hh] = slots[e,:] . w1[e,hh,:] + b1 (wave per output) ----------------
__global__ void k_h(const float* __restrict__ slots, const float* __restrict__ w1,
                    const float* __restrict__ b1, float* __restrict__ h_pre) {
    const int lane = threadIdx.x & 31;
    const int wid = (blockIdx.x * blockDim.x + threadIdx.x) >> 5;   // 32768 waves
    const int e = wid >> 11, hh = wid & (HDIM - 1);
    const float4* wrow = (const float4*)(w1 + ((size_t)e * HDIM + hh) * DIM);
    const float4* srow = (const float4*)(slots + (size_t)e * DIM);
    float acc = 0.f;
    #pragma unroll
    for (int i = 0; i < 8; ++i) {
        float4 wv = wrow[i * 32 + lane];
        float4 sv = srow[i * 32 + lane];
        acc += wv.x * sv.x + wv.y * sv.y + wv.z * sv.z + wv.w * sv.w;
    }
    for (int off = 16; off > 0; off >>= 1) acc += __shfl_down(acc, off, 32);
    if (lane == 0)
        h_pre[(size_t)e * HDIM + hh] = acc + b1[(size_t)e * HDIM + hh];
}

// ---------------- LayerNorm (biased var) + exact-erf GELU ----------------
__global__ void k_ln_gelu(const float* __restrict__ h_pre, const float* __restrict__ ln_g,
                          const float* __restrict__ ln_b, float* __restrict__ h_act) {
    __shared__ float s1[256], s2[256];
    const int e = blockIdx.x, tid = threadIdx.x;
    const float* row = h_pre + (size_t)e * HDIM;
    float v[8], sm = 0.f, sq = 0.f;
    #pragma unroll
    for (int i = 0; i < 8; ++i) {
        v[i] = row[i * 256 + tid];
        sm += v[i]; sq += v[i] * v[i];
    }
    s1[tid] = sm; s2[tid] = sq; __syncthreads();
    for (int off = 128; off > 0; off >>= 1) {
        if (tid < off) { s1[tid] += s1[tid + off]; s2[tid] += s2[tid + off]; }
        __syncthreads();
    }
    const float mu = s1[0] * (1.f / HDIM);
    const float var = s2[0] * (1.f / HDIM) - mu * mu;
    const float rs = rsqrtf(var + 1e-5f);
    #pragma unroll
    for (int i = 0; i < 8; ++i) {
        int idx = i * 256 + tid;
        float t = (v[i] - mu) * rs * ln_g[(size_t)e * HDIM + idx] + ln_b[(size_t)e * HDIM + idx];
        h_act[(size_t)e * HDIM + idx] = 0.5f * t * (1.f + erff(t * 0.70710678118654752f));
    }
}

// ---------------- slot_out[e,d] = h_act[e,:] . w2[e,d,:] + b2 (wave per output) ----------------
__global__ void k_slot_out(const float* __restrict__ h_act, const float* __restrict__ w2,
                           const float* __restrict__ b2, float* __restrict__ slot_out) {
    const int lane = threadIdx.x & 31;
    const int wid = (blockIdx.x * blockDim.x + threadIdx.x) >> 5;   // 16384 waves
    const int e = wid >> 10, d = wid & (DIM - 1);
    const float4* wrow = (const float4*)(w2 + ((size_t)e * DIM + d) * HDIM);
    const float4* hrow = (const float4*)(h_act + (size_t)e * HDIM);
    float acc = 0.f;
    #pragma unroll
    for (int i = 0; i < 16; ++i) {
        float4 wv = wrow[i * 32 + lane];
        float4 hv = hrow[i * 32 + lane];
        acc += wv.x * hv.x + wv.y * hv.y + wv.z * hv.z + wv.w * hv.w;
    }
    for (int off = 16; off > 0; off >>= 1) acc += __shfl_down(acc, off, 32);
    if (lane == 0)
        slot_out[(size_t)e * DIM + d] = acc + b2[(size_t)e * DIM + d];
}

// ---------------- out = combine @ slot_out  (WMMA f32 16x16x4, K=16 experts) ----------------
__global__ void k_out_wmma(const float* __restrict__ combine, const float* __restrict__ slot_out,
                           float* __restrict__ out) {
    const int lane = threadIdx.x & 31;
    const int wave = threadIdx.x >> 5;
    const int wid = blockIdx.x * 8 + wave;       // 32768 waves = 512 x 64 tiles
    const int t0 = (wid >> 6) * 16;
    const int d0 = (wid & 63) * 16;
    const int n  = lane & 15;
    const int kk = (lane >> 4) << 1;
    v8f c = {};
    #pragma unroll
    for (int k = 0; k < NEXP; k += 4) {
        v2f a, b;
        a[0] = combine[(size_t)(t0 + n) * NEXP + k + kk];
        a[1] = combine[(size_t)(t0 + n) * NEXP + k + kk + 1];
        b[0] = slot_out[(size_t)(k + kk) * DIM + d0 + n];
        b[1] = slot_out[(size_t)(k + kk + 1) * DIM + d0 + n];
        c = __builtin_amdgcn_wmma_f32_16x16x4_f32(false, a, false, b, (short)0, c,
                                                  false, false);
    }
    const int rbase = t0 + ((lane >> 4) << 3);
    #pragma unroll
    for (int i = 0; i < 8; ++i)
        out[(size_t)(rbase + i) * DIM + d0 + n] = c[i];
}

// ---------------- aux loss + entropy scalars ----------------
__global__ void k_scalars(const float* __restrict__ rw, const float* __restrict__ usage,
                          const float* __restrict__ ent_acc, float* __restrict__ out_tail) {
    __shared__ float s[256];
    __shared__ float mm[NEXP];
    const int tid = threadIdx.x;
    const int e = tid >> 4, seg = tid & 15;
    const float* row = rw + (size_t)e * DIM + seg * 64;
    float p = 0.f;
    for (int j = 0; j < 64; ++j) p += row[j];
    s[tid] = p; __syncthreads();
    if (tid < NEXP) {
        float t = 0.f;
        for (int j = 0; j < 16; ++j) t += s[tid * 16 + j];
        mm[tid] = t * (1.f / DIM);
    }
    __syncthreads();
    if (tid == 0) {
        float mx = mm[0];
        for (int i = 1; i < NEXP; ++i) mx = fmaxf(mx, mm[i]);
        float pe[NEXP], se = 0.f;
        for (int i = 0; i < NEXP; ++i) { pe[i] = expf(mm[i] - mx); se += pe[i]; }
        float loss = 0.f;
        for (int i = 0; i < NEXP; ++i)
            loss += (pe[i] / se) * (usage[i] * (1.f / T_TOK));
        out_tail[0] = (float)NEXP * loss;
        out_tail[1] = ent_acc[0] * (1.f / T_TOK);
    }
}

extern "C" void kernel_launch(void* const* d_in, const int* in_sizes, int n_in,
                              void* d_out, int out_size, void* d_ws, size_t ws_size,
                              hipStream_t stream) {
    const float* x    = (const float*)d_in[0];
    const float* rw   = (const float*)d_in[1];
    const float* w1   = (const float*)d_in[2];
    const float* b1   = (const float*)d_in[3];
    const float* ln_g = (const float*)d_in[4];
    const float* ln_b = (const float*)d_in[5];
    const float* w2   = (const float*)d_in[6];
    const float* b2   = (const float*)d_in[7];
    float* out = (float*)d_out;
    float* ws  = (float*)d_ws;

    float* logits   = ws + WS_LOGITS;
    float* combine  = ws + WS_COMBINE;
    float* colmax   = ws + WS_COLMAX;
    float* colsum   = ws + WS_COLSUM;
    float* usage    = ws + WS_USAGE;
    float* ent      = ws + WS_ENT;
    float* slots    = ws + WS_SLOTS;
    float* h_pre    = ws + WS_HPRE;
    float* h_act    = ws + WS_HACT;
    float* slot_out = ws + WS_SLOTOUT;

    k_init<<<64, 256, 0, stream>>>(slots, usage, ent);
    k_logits_wmma<<<64, 256, 0, stream>>>(x, rw, logits);
    k_colstats<<<NEXP, 256, 0, stream>>>(logits, colmax, colsum);
    k_combine<<<T_TOK / 256, 256, 0, stream>>>(logits, combine, usage, ent);
    k_slots<<<dim3(16, NEXP), 256, 0, stream>>>(x, logits, colmax, colsum, slots);
    k_h<<<4096, 256, 0, stream>>>(slots, w1, b1, h_pre);
    k_ln_gelu<<<NEXP, 256, 0, stream>>>(h_pre, ln_g, ln_b, h_act);
    k_slot_out<<<2048, 256, 0, stream>>>(h_act, w2, b2, slot_out);
    k_out_wmma<<<4096, 256, 0, stream>>>(combine, slot_out, out);
    k_scalars<<<1, 256, 0, stream>>>(rw, usage, ent, out + (size_t)T_TOK * DIM);
}